// BitwiseHashing_87497073754448
// MI455X (gfx1250) — compile-verified
//
#include <hip/hip_runtime.h>
#include <math.h>

// Problem constants (from reference): x[L,B,D], W[K,D], b[K] -> out[B,K]
constexpr int Lc = 12;
constexpr int Bc = 8192;
constexpr int Dc = 1024;
constexpr int Kc = 64;

constexpr int M_TILE  = 32;   // rows of B per block
constexpr int D_CHUNK = 256;  // D columns staged in LDS per pass
constexpr int PITCH   = 260;  // D_CHUNK + 4 floats pad: 16B-aligned rows, bank-conflict-free
constexpr int NTHREADS = 256; // 8 waves (wave32): one 16x16 C tile per wave

typedef __attribute__((ext_vector_type(2))) float v2f;
typedef __attribute__((ext_vector_type(8))) float v8f;

__global__ __launch_bounds__(NTHREADS)
void fused_mean_gemm_tanh(const float* __restrict__ x,
                          const float* __restrict__ W,
                          const float* __restrict__ bias,
                          float* __restrict__ out)
{
    __shared__ float xm_lds[M_TILE * PITCH];  // mean-over-L chunk [32][260]
    __shared__ float w_lds[Kc * PITCH];       // W chunk           [64][260]

    const int tid   = threadIdx.x;
    const int lane  = tid & 31;
    const int wv    = tid >> 5;        // wave id: 0..7
    const int hi    = lane >> 4;       // lane half (selects K=0,1 vs K=2,3 of frag)
    const int ln    = lane & 15;
    const int b0    = blockIdx.x * M_TILE;

    const int ntile = wv & 3;          // which 16-col tile of K (0..3)
    const int mrow  = (wv >> 2) * 16;  // which 16-row strip of the M tile (0 or 16)

    v8f c0 = {};                       // one 16x16 f32 accumulator tile per wave

    for (int d0 = 0; d0 < Dc; d0 += D_CHUNK) {
        // ---- Stage W chunk: 64 rows x 64 float4 ----
        for (int e = tid; e < Kc * (D_CHUNK / 4); e += NTHREADS) {
            const int row = e >> 6;
            const int c4  = (e & 63) << 2;
            const float4 w4 = *(const float4*)(W + (size_t)row * Dc + d0 + c4);
            *(float4*)(w_lds + row * PITCH + c4) = w4;
        }
        // ---- Stream x over L, accumulate mean chunk: 32 rows x 64 float4 ----
        for (int e = tid; e < M_TILE * (D_CHUNK / 4); e += NTHREADS) {
            const int row = e >> 6;
            const int c4  = (e & 63) << 2;
            const float* p = x + (size_t)(b0 + row) * Dc + d0 + c4;
            float4 acc = {0.f, 0.f, 0.f, 0.f};
#pragma unroll
            for (int l = 0; l < Lc; ++l) {
                const float4 v = *(const float4*)(p + (size_t)l * Bc * Dc);
                acc.x += v.x; acc.y += v.y; acc.z += v.z; acc.w += v.w;
            }
            const float s = 1.0f / (float)Lc;
            acc.x *= s; acc.y *= s; acc.z *= s; acc.w *= s;
            *(float4*)(xm_lds + row * PITCH + c4) = acc;
        }
        __syncthreads();

        // ---- GEMM on the chunk via V_WMMA_F32_16X16X4_F32 ----
        // A frag (16x4 f32): lane m<16 holds (K=0,K=1); lane m+16 holds (K=2,K=3)
        // B frag (4x16 f32): mirrored, from W rows (n = ln within this wave's n-tile)
        const float* arow = xm_lds + (mrow + ln) * PITCH + 2 * hi;
        const float* brow = w_lds  + (ntile * 16 + ln) * PITCH + 2 * hi;
#pragma unroll 8
        for (int d = 0; d < D_CHUNK; d += 4) {
            const v2f a = *(const v2f*)(arow + d);
            const v2f b = *(const v2f*)(brow + d);
            c0 = __builtin_amdgcn_wmma_f32_16x16x4_f32(false, a, false, b,
                                                       (short)0, c0, false, false);
        }
        __syncthreads();
    }

    // ---- Epilogue: bias + tanh, coalesced stores ----
    // C/D f32 16x16 layout: VGPR r, lanes 0-15 -> M=r, lanes 16-31 -> M=r+8; N=ln
    const int k   = ntile * 16 + ln;
    const float bk = bias[k];
#pragma unroll
    for (int r = 0; r < 8; ++r) {
        const int bb = b0 + mrow + r + 8 * hi;
        out[(size_t)bb * Kc + k] = tanhf(c0[r] + bk);
    }
}

extern "C" void kernel_launch(void* const* d_in, const int* in_sizes, int n_in,
                              void* d_out, int out_size, void* d_ws, size_t ws_size,
                              hipStream_t stream) {
    const float* x    = (const float*)d_in[0];
    const float* W    = (const float*)d_in[1];
    const float* bias = (const float*)d_in[2];
    float* out        = (float*)d_out;

    dim3 grid(Bc / M_TILE);   // 256 blocks
    dim3 block(NTHREADS);     // 8 waves (wave32)
    fused_mean_gemm_tanh<<<grid, block, 0, stream>>>(x, W, bias, out);
}